// MambaBlock_80994493268305
// MI455X (gfx1250) — compile-verified
//
#include <hip/hip_runtime.h>
#include <hip/hip_bf16.h>
#include <math.h>

// Mamba block forward for MI455X (gfx1250), fp32 WMMA path, register-blocked,
// compile-time leading dimensions so all inner-loop loads use immediate offsets.
#define BB   2
#define LL   2048
#define DM   1024
#define DI   2048      // d_inner
#define NST  16        // d_state
#define DTR  64        // dt_rank
#define MM   (BB*LL)   // 4096 rows

typedef __attribute__((ext_vector_type(2))) float v2f;
typedef __attribute__((ext_vector_type(8))) float v8f;

// ---------------------------------------------------------------------------
// fp32 WMMA GEMM, register blocked: each wave computes a 64(M) x 32(N) patch
// of C = A[M,K] @ B[K,N] (row-major). Per K-step of 4: 4 b64 (A) + 4 b32 (B)
// loads vs 8 WMMAs. LDA/LDB/K are compile-time -> all loads are
// base + immediate-offset; only 6 base pointers bumped once per 32-K chunk.
// EPI: 0 = plain store, 1 = softplus(c + bias[col]).
// ---------------------------------------------------------------------------
template <int EPI, int LDA, int LDB, int K>
__global__ void wmma_gemm_f32_blk(const float* __restrict__ A,
                                  const float* __restrict__ B,
                                  float* __restrict__ C,
                                  const float* __restrict__ bias,
                                  int M, int N, int ldc) {
    const int wavesPerBlock = blockDim.x >> 5;
    const int tile   = blockIdx.x * wavesPerBlock + (threadIdx.x >> 5);
    const int tilesN = N >> 5;                 // 32-wide patches
    const int tm = tile / tilesN;              // 64-tall patches
    const int tn = tile - tm * tilesN;
    if (tm >= (M >> 6)) return;                // never taken (exact grids)

    const int lane = threadIdx.x & 31;
    const int lo   = lane & 15;                // row (A) / col (B,C) in 16-tile
    const int hi   = lane >> 4;                // selects K pair {0,1} vs {2,3}

    const int rowBase = tm << 6;
    const int colBase = tn << 5;

    // A frag r base: A[rowBase+16r+lo][2*hi]  (pairs are contiguous -> b64)
    const float* a0p = A + (size_t)(rowBase +  0 + lo) * LDA + 2 * hi;
    const float* a1p = A + (size_t)(rowBase + 16 + lo) * LDA + 2 * hi;
    const float* a2p = A + (size_t)(rowBase + 32 + lo) * LDA + 2 * hi;
    const float* a3p = A + (size_t)(rowBase + 48 + lo) * LDA + 2 * hi;
    // B frag s base: B[2*hi][colBase+16s+lo]  (pairs strided by LDB -> 2 b32)
    const float* b0p = B + (size_t)(2 * hi) * LDB + (colBase +  0 + lo);
    const float* b1p = B + (size_t)(2 * hi) * LDB + (colBase + 16 + lo);

    v8f c00 = {}, c01 = {}, c10 = {}, c11 = {};
    v8f c20 = {}, c21 = {}, c30 = {}, c31 = {};

    for (int k0 = 0; k0 < K; k0 += 32) {
        // Prefetch the chunk two ahead (speculative; dropped if OOB).
        __builtin_prefetch(a0p + 64, 0, 1);
        __builtin_prefetch(a1p + 64, 0, 1);
        __builtin_prefetch(a2p + 64, 0, 1);
        __builtin_prefetch(a3p + 64, 0, 1);
        __builtin_prefetch(b0p + (size_t)64 * LDB, 0, 1);
        __builtin_prefetch(b1p + (size_t)64 * LDB, 0, 1);

#pragma unroll
        for (int kk = 0; kk < 32; kk += 4) {
            v2f a0 = *(const v2f*)(a0p + kk);
            v2f a1 = *(const v2f*)(a1p + kk);
            v2f a2 = *(const v2f*)(a2p + kk);
            v2f a3 = *(const v2f*)(a3p + kk);
            v2f b0, b1;
            b0.x = b0p[(size_t)kk * LDB]; b0.y = b0p[(size_t)(kk + 1) * LDB];
            b1.x = b1p[(size_t)kk * LDB]; b1.y = b1p[(size_t)(kk + 1) * LDB];

            c00 = __builtin_amdgcn_wmma_f32_16x16x4_f32(false, a0, false, b0, (short)0, c00, false, false);
            c01 = __builtin_amdgcn_wmma_f32_16x16x4_f32(false, a0, false, b1, (short)0, c01, false, false);
            c10 = __builtin_amdgcn_wmma_f32_16x16x4_f32(false, a1, false, b0, (short)0, c10, false, false);
            c11 = __builtin_amdgcn_wmma_f32_16x16x4_f32(false, a1, false, b1, (short)0, c11, false, false);
            c20 = __builtin_amdgcn_wmma_f32_16x16x4_f32(false, a2, false, b0, (short)0, c20, false, false);
            c21 = __builtin_amdgcn_wmma_f32_16x16x4_f32(false, a2, false, b1, (short)0, c21, false, false);
            c30 = __builtin_amdgcn_wmma_f32_16x16x4_f32(false, a3, false, b0, (short)0, c30, false, false);
            c31 = __builtin_amdgcn_wmma_f32_16x16x4_f32(false, a3, false, b1, (short)0, c31, false, false);
        }
        a0p += 32; a1p += 32; a2p += 32; a3p += 32;
        b0p += (size_t)32 * LDB; b1p += (size_t)32 * LDB;
    }

    // C/D layout: VGPR i -> row = i + 8*hi, col = lo (within each 16x16 tile)
    v8f* acc[4][2] = {{&c00, &c01}, {&c10, &c11}, {&c20, &c21}, {&c30, &c31}};
#pragma unroll
    for (int r = 0; r < 4; ++r) {
#pragma unroll
        for (int s = 0; s < 2; ++s) {
            const int cc = colBase + 16 * s + lo;
            float bv = (EPI == 1) ? bias[cc] : 0.f;
#pragma unroll
            for (int i = 0; i < 8; ++i) {
                const int rr = rowBase + 16 * r + i + 8 * hi;
                float v = (*acc[r][s])[i];
                if (EPI == 1) {
                    v += bv;
                    v = (v > 20.f) ? v : log1pf(expf(v));   // softplus
                }
                C[(size_t)rr * ldc + cc] = v;
            }
        }
    }
}

// ---------------------------------------------------------------------------
// Causal depthwise conv1d (width 4) + bias + SiLU.
// ---------------------------------------------------------------------------
__global__ void conv_silu_kernel(const float* __restrict__ xz,
                                 const float* __restrict__ conv_w,
                                 const float* __restrict__ conv_b,
                                 float* __restrict__ x_conv) {
    const int idx = blockIdx.x * blockDim.x + threadIdx.x;   // over MM*DI
    const int d  = idx & (DI - 1);
    const int bl = idx >> 11;          // DI = 2048 = 2^11
    const int l  = bl & (LL - 1);
    const int b  = bl >> 11;           // LL = 2048 = 2^11

    float acc = conv_b[d];
#pragma unroll
    for (int k = 0; k < 4; ++k) {
        const int lt = l - 3 + k;
        if (lt >= 0) {
            acc += xz[((size_t)(b * LL + lt)) * (2 * DI) + d] * conv_w[d * 4 + k];
        }
    }
    x_conv[(size_t)bl * DI + d] = acc / (1.f + expf(-acc));  // silu
}

// ---------------------------------------------------------------------------
// Selective scan. One thread per (b, d); h[16] in registers; b block-uniform.
// Fuses skip (D_param) and gate (silu(z)).
// ---------------------------------------------------------------------------
__global__ void scan_kernel(const float* __restrict__ xz,
                            const float* __restrict__ x_conv,
                            const float* __restrict__ ssm,
                            const float* __restrict__ delta,
                            const float* __restrict__ A_log,
                            const float* __restrict__ D_param,
                            float* __restrict__ y_gated) {
    const int b = blockIdx.x >> 3;                       // 8 blocks per batch
    const int d = ((blockIdx.x & 7) << 8) + threadIdx.x; // 0..DI-1

    float Ac[NST];
#pragma unroll
    for (int n = 0; n < NST; ++n) Ac[n] = -expf(A_log[d * NST + n]);
    const float Dp = D_param[d];

    float h[NST];
#pragma unroll
    for (int n = 0; n < NST; ++n) h[n] = 0.f;

    const size_t rowBase = (size_t)b * LL;
    for (int t = 0; t < LL; ++t) {
        const size_t r = rowBase + t;
        const float dt = delta[r * DI + d];
        const float xv = x_conv[r * DI + d];
        const float* bc = ssm + r * (DTR + 2 * NST);     // [dtr | B | C] row

        float y = 0.f;
#pragma unroll
        for (int n = 0; n < NST; ++n) {
            const float dA = expf(dt * Ac[n]);
            h[n] = dA * h[n] + (dt * bc[DTR + n]) * xv;
            y += h[n] * bc[DTR + NST + n];
        }

        const float zv = xz[r * (2 * DI) + DI + d];
        const float gate = zv / (1.f + expf(-zv));       // silu(z)
        y_gated[r * DI + d] = (y + xv * Dp) * gate;
    }
}

// ---------------------------------------------------------------------------
extern "C" void kernel_launch(void* const* d_in, const int* in_sizes, int n_in,
                              void* d_out, int out_size, void* d_ws, size_t ws_size,
                              hipStream_t stream) {
    const float* x       = (const float*)d_in[0];
    const float* W_in    = (const float*)d_in[1];
    const float* conv_w  = (const float*)d_in[2];
    const float* conv_b  = (const float*)d_in[3];
    const float* W_x     = (const float*)d_in[4];
    const float* W_dt    = (const float*)d_in[5];
    const float* b_dt    = (const float*)d_in[6];
    const float* A_log   = (const float*)d_in[7];
    const float* D_param = (const float*)d_in[8];
    const float* W_out   = (const float*)d_in[9];
    float* out = (float*)d_out;

    // Workspace layout (bytes)
    char* ws = (char*)d_ws;
    float* xz      = (float*)(ws);                        // [MM, 2*DI]   64 MB
    float* x_conv  = (float*)(ws + (size_t)67108864);     // [MM, DI]     32 MB
    float* ssm     = (float*)(ws + (size_t)100663296);    // [MM, 96]     1.5 MB
    float* delta   = (float*)(ws + (size_t)102236160);    // [MM, DI]     32 MB
    float* y_g     = (float*)(ws + (size_t)135790592);    // [MM, DI]     32 MB

    // 1) xz = x @ W_in                (4096 x 1024 x 4096)
    {
        const int waves = (MM / 64) * ((2 * DI) / 32);    // 8192
        wmma_gemm_f32_blk<0, DM, 2 * DI, DM><<<waves / 8, 256, 0, stream>>>(
            x, W_in, xz, nullptr, MM, 2 * DI, 2 * DI);
    }
    // 2) causal conv + SiLU
    conv_silu_kernel<<<(MM * DI) / 256, 256, 0, stream>>>(xz, conv_w, conv_b, x_conv);
    // 3) ssm = x_conv @ W_x           (4096 x 2048 x 96)
    {
        const int waves = (MM / 64) * (96 / 32);          // 192
        wmma_gemm_f32_blk<0, DI, 96, DI><<<waves / 8, 256, 0, stream>>>(
            x_conv, W_x, ssm, nullptr, MM, 96, 96);
    }
    // 4) delta = softplus(ssm[:, :64] @ W_dt + b_dt)   (4096 x 64 x 2048)
    {
        const int waves = (MM / 64) * (DI / 32);          // 4096
        wmma_gemm_f32_blk<1, 96, DI, DTR><<<waves / 8, 256, 0, stream>>>(
            ssm, W_dt, delta, b_dt, MM, DI, DI);
    }
    // 5) selective scan + skip + gate
    scan_kernel<<<BB * (DI / 256), 256, 0, stream>>>(
        xz, x_conv, ssm, delta, A_log, D_param, y_g);
    // 6) out = y_gated @ W_out        (4096 x 2048 x 1024)
    {
        const int waves = (MM / 64) * (DM / 32);          // 2048
        wmma_gemm_f32_blk<0, DI, DM, DI><<<waves / 8, 256, 0, stream>>>(
            y_g, W_out, out, nullptr, MM, DM, DM);
    }
}